// Evaluator_40870908788848
// MI455X (gfx1250) — compile-verified
//
#include <hip/hip_runtime.h>
#include <stdint.h>

namespace {
constexpr int H    = 2048;
constexpr int W    = 2048;
constexpr int PAD  = 4;                 // KERNEL_SIZE=9 -> pad 4, edge mode
constexpr int PW   = W + 2 * PAD;       // 2056 padded width/height
constexpr int TILE = 32;
constexpr int TDIM = TILE + 2 * PAD;    // 40
constexpr int HW   = H * W;
constexpr int NOFF = 80;                // 9*9 - 1 offsets
constexpr int NRING = 14;               // distinct distances

typedef unsigned int v4u __attribute__((ext_vector_type(4)));
typedef int          v8i __attribute__((ext_vector_type(8)));
typedef int          v4i __attribute__((ext_vector_type(4)));

// Offsets sorted by DESCENDING distance, grouped into equal-distance rings.
// Iterating the conditional-select chain in this order makes the final value
// the minimum mismatching distance (last writer = smallest ring).
struct Table {
    signed char dy[NOFF];
    signed char dx[NOFF];
    signed char ring[NOFF];
    short       rep[NRING];   // flat dker index of one representative per ring
};
constexpr Table build_table() {
    Table t{};
    const int rd2[NRING] = {32, 25, 20, 18, 17, 16, 13, 10, 9, 8, 5, 4, 2, 1};
    int n = 0;
    for (int r = 0; r < NRING; ++r) {
        bool found = false;
        for (int dy = 0; dy <= 4; ++dy)
            for (int dx = 0; dx <= 4; ++dx)
                if (!found && (dy * dy + dx * dx) == rd2[r]) {
                    t.rep[r] = (short)((dy + 4) * 9 + (dx + 4));
                    found = true;
                }
        for (int dy = -4; dy <= 4; ++dy)
            for (int dx = -4; dx <= 4; ++dx) {
                if (dy == 0 && dx == 0) continue;
                if (dy * dy + dx * dx == rd2[r]) {
                    t.dy[n] = (signed char)dy;
                    t.dx[n] = (signed char)dx;
                    t.ring[n] = (signed char)r;
                    ++n;
                }
            }
    }
    return t;
}
} // namespace

// ---------------------------------------------------------------------------
// Pass 1: fw = sum over 5 channels; mask -> interior of padded array
// ---------------------------------------------------------------------------
__global__ void k_sum_mask(const float* __restrict__ target,
                           float* __restrict__ fw,
                           float* __restrict__ padded) {
    int idx = blockIdx.x * blockDim.x + threadIdx.x;
    if (idx >= HW) return;
    float s = target[idx]
            + target[idx + 1 * HW]
            + target[idx + 2 * HW]
            + target[idx + 3 * HW]
            + target[idx + 4 * HW];
    fw[idx] = s;
    int y = idx >> 11;          // /2048
    int x = idx & (W - 1);
    padded[(size_t)(y + PAD) * PW + (x + PAD)] = (s > 0.0f) ? 1.0f : 0.0f;
}

// ---------------------------------------------------------------------------
// Pass 2: edge-replicate the 4-wide border of the padded array; init min/max
// ---------------------------------------------------------------------------
__global__ void k_pad(float* __restrict__ padded, unsigned* __restrict__ minmax) {
    int idx = blockIdx.x * blockDim.x + threadIdx.x;
    if (idx == 0) {
        minmax[0] = 0x7F800000u;   // +inf bits (min accumulator)
        minmax[1] = 0u;            // 0.0f bits (max accumulator; weights >= 0)
    }
    if (idx >= PW * PW) return;
    int r = idx / PW;
    int c = idx - r * PW;
    if (r >= PAD && r < PW - PAD && c >= PAD && c < PW - PAD) return; // interior done
    int rr = min(max(r, PAD), PW - PAD - 1);
    int cc = min(max(c, PAD), PW - PAD - 1);
    padded[idx] = padded[(size_t)rr * PW + cc];
}

// ---------------------------------------------------------------------------
// Pass 3: TDM-staged 9x9 stencil -> unnormalized weights + global min/max
// blockDim = (32, 8); each block = 32x32 tile; thread handles 4 CONSECUTIVE
// rows so its four 9x9 windows share 12x9 unique LDS loads (compiler CSE).
// ---------------------------------------------------------------------------
__global__ __launch_bounds__(256) void k_stencil(const float* __restrict__ padded,
                                                 const float* __restrict__ fw,
                                                 const float* __restrict__ dker,
                                                 float* __restrict__ wout,
                                                 unsigned* __restrict__ minmax) {
    __shared__ float tile[TDIM * TDIM];   // 40x40 halo tile (6400 B)
    __shared__ float red[16];             // per-wave min(8) / max(8)

    const int tx = threadIdx.x;           // 0..31
    const int ty = threadIdx.y;           // 0..7
    const int flat = ty * 32 + tx;
    const int bx = blockIdx.x, by = blockIdx.y;

    // 14 ring distances: uniform constant-index loads -> scalar loads
    constexpr Table TAB = build_table();
    float RD[NRING];
    #pragma unroll
    for (int r = 0; r < NRING; ++r) RD[r] = dker[TAB.rep[r]];

    // ---- Tensor Data Mover: DMA the 40x40 tile (always fully in-bounds of
    // the 2056x2056 padded array) from global into LDS. Issued by wave 0 only.
    if (ty == 0) {
        const float* gsrc = padded + (size_t)(by * TILE) * PW + (size_t)(bx * TILE);
        uint64_t gaddr = (uint64_t)(uintptr_t)gsrc;
        unsigned lds_addr = (unsigned)(uintptr_t)&tile[0];   // low 32 bits = LDS byte offset

        v4u g0;
        g0.x = 1u;                                            // count=1, user descriptor
        g0.y = lds_addr;                                      // lds_addr [63:32]
        g0.z = (unsigned)gaddr;                               // global_addr [95:64]
        g0.w = ((unsigned)(gaddr >> 32) & 0x01FFFFFFu)        // global_addr [120:96]
             | (2u << 30);                                    // type=2 ("image")

        v8i g1;
        g1[0] = (int)(2u << 16);                              // data_size=2 (4 bytes)
        g1[1] = (int)((unsigned)PW << 16);                    // tensor_dim0[15:0]
        g1[2] = (int)((unsigned)PW << 16);                    // dim0[31:16]=0 | tensor_dim1[15:0]
        g1[3] = (int)((unsigned)TDIM << 16);                  // dim1[31:16]=0 | tile_dim0=40
        g1[4] = TDIM;                                         // tile_dim1=40, tile_dim2=0
        g1[5] = PW;                                           // tensor_dim0_stride[31:0]
        g1[6] = 0;                                            // stride[47:32]=0, dim1_stride lo=0
        g1[7] = 0;

        v4i gz4 = {0, 0, 0, 0};
        v8i gz8 = {0, 0, 0, 0, 0, 0, 0, 0};
        __builtin_amdgcn_tensor_load_to_lds(g0, g1, gz4, gz4, gz8, 0);
        __builtin_amdgcn_s_wait_tensorcnt(0);
    }
    __syncthreads();   // publish LDS tile to all 8 waves

    const float INF = __builtin_inff();
    const int ty4 = ty * 4;                         // first of 4 consecutive rows
    const float* tbase = &tile[(ty4 + PAD) * TDIM + (tx + PAD)];

    float c[4], md[4];
    #pragma unroll
    for (int p = 0; p < 4; ++p) {
        c[p]  = tbase[p * TDIM];
        md[p] = INF;
    }

    // Descending-distance conditional-select chain == min over mismatches.
    #pragma unroll
    for (int k = 0; k < NOFF; ++k) {
        const int dy = TAB.dy[k], dx = TAB.dx[k];
        const float D = RD[(int)TAB.ring[k]];
        #pragma unroll
        for (int p = 0; p < 4; ++p) {
            float v = tbase[(p + dy) * TDIM + dx];  // 108 unique loads, CSE'd
            md[p] = (v != c[p]) ? D : md[p];
        }
    }

    float mn = INF, mx = 0.0f;
    #pragma unroll
    for (int p = 0; p < 4; ++p) {
        float contour = (md[p] < INF) ? (1.0f / (md[p] + 1e-10f)) : 0.0f;
        const int gy = by * TILE + ty4 + p;
        const int gx = bx * TILE + tx;
        float w = fw[(size_t)gy * W + gx] + contour;
        w = w * w;
        wout[(size_t)gy * W + gx] = w;              // coalesced: wave = one row
        mn = fminf(mn, w);
        mx = fmaxf(mx, w);
    }

    // wave32 reduction, then cross-wave via LDS, then one atomic pair per block
    #pragma unroll
    for (int off = 16; off > 0; off >>= 1) {
        mn = fminf(mn, __shfl_xor(mn, off, 32));
        mx = fmaxf(mx, __shfl_xor(mx, off, 32));
    }
    if ((flat & 31) == 0) {
        red[flat >> 5]       = mn;
        red[8 + (flat >> 5)] = mx;
    }
    __syncthreads();
    if (flat == 0) {
        float m0 = red[0], m1 = red[8];
        #pragma unroll
        for (int k = 1; k < 8; ++k) {
            m0 = fminf(m0, red[k]);
            m1 = fmaxf(m1, red[8 + k]);
        }
        // all weights >= 0 -> IEEE bits are monotone as unsigned ints
        atomicMin(&minmax[0], __float_as_uint(m0));
        atomicMax(&minmax[1], __float_as_uint(m1));
    }
}

// ---------------------------------------------------------------------------
// Pass 4: normalize in place and apply mask
// ---------------------------------------------------------------------------
__global__ void k_norm(float* __restrict__ out,
                       const float* __restrict__ padded,
                       const unsigned* __restrict__ minmax) {
    int idx = blockIdx.x * blockDim.x + threadIdx.x;
    if (idx >= HW) return;
    float mnv = __uint_as_float(minmax[0]);
    float mxv = __uint_as_float(minmax[1]);
    int y = idx >> 11;
    int x = idx & (W - 1);
    float mask = padded[(size_t)(y + PAD) * PW + (x + PAD)];
    float w = out[idx];
    out[idx] = (w - mnv) / (mxv - mnv + 1e-10f) * mask;
}

// ---------------------------------------------------------------------------
extern "C" void kernel_launch(void* const* d_in, const int* in_sizes, int n_in,
                              void* d_out, int out_size, void* d_ws, size_t ws_size,
                              hipStream_t stream) {
    const float* target = (const float*)d_in[0];     // (5, 2048, 2048) f32
    const float* dker   = (const float*)d_in[1];     // (9, 9) f32
    float* out = (float*)d_out;                      // (2048, 2048) f32

    // workspace layout
    float*    padded = (float*)d_ws;                 // PW*PW floats (~16.9 MB)
    float*    fw     = padded + (size_t)PW * PW;     // H*W floats  (~16.8 MB)
    unsigned* minmax = (unsigned*)(fw + (size_t)HW); // 2 uints

    (void)in_sizes; (void)n_in; (void)out_size; (void)ws_size;

    k_sum_mask<<<(HW + 255) / 256, 256, 0, stream>>>(target, fw, padded);
    k_pad<<<(PW * PW + 255) / 256, 256, 0, stream>>>(padded, minmax);
    k_stencil<<<dim3(W / TILE, H / TILE), dim3(32, 8), 0, stream>>>(padded, fw, dker, out, minmax);
    k_norm<<<(HW + 255) / 256, 256, 0, stream>>>(out, padded, minmax);
}